// MambaClassifier_2534030705270
// MI455X (gfx1250) — compile-verified
//
#include <hip/hip_runtime.h>

// ---------------------------------------------------------------------------
// Mamba classifier for MI455X (gfx1250, wave32).
// fp32 end-to-end; all GEMMs use V_WMMA_F32_16X16X4_F32.
// Fragment setup is branch-free: clamped addresses + minimal cndmask selects.
// ---------------------------------------------------------------------------

typedef __attribute__((ext_vector_type(2))) float v2f;
typedef __attribute__((ext_vector_type(8))) float v8f;

#define B_SZ    8
#define L_SZ    2048
#define MTOT    (B_SZ * L_SZ)     // 16384
#define MT      (MTOT / 16)       // 1024 M-tiles
#define DM      6
#define ED      48
#define NST     32
#define DCONV   16
#define XZ_N    96                // 2*ED, stride of xz buffer
#define DBL_N   80                // 65 padded to 5 N-tiles of 16
#define EPS_    1e-5f

__device__ __forceinline__ float fast_sigmoid(float x) {
    return 1.0f / (1.0f + __expf(-x));
}

// ---------------------------------------------------------------------------
// K1: RMSNorm + in_proj.  xz[m, 0:96] = rmsnorm(h[m,0:6]) @ in_proj_w.T
// GEMM M=16384 N=96 K=6 via wmma f32 16x16x4 (2 K-steps).
// K=6,7 padding handled by zeroing ONLY the B fragment on the upper half-wave
// (A slot values are then multiplied by zero, so they can be arbitrary).
// ---------------------------------------------------------------------------
__global__ void k_norm_inproj(const float* __restrict__ hin,   // (M, 6)
                              const float* __restrict__ w,     // (96, 6)
                              const float* __restrict__ normw, // (6)
                              float* __restrict__ xz)          // (M, 96)
{
    const int gw   = blockIdx.x * 4 + (threadIdx.x >> 5);   // global wave id
    const int mt   = gw / 6;
    const int nt   = gw % 6;
    const int t    = threadIdx.x & 31;
    const int l16  = t & 15;
    const bool hi  = (t >> 4) != 0;     // upper half-wave (K offset +2)

    // --- A-side: normalized row (computed redundantly by both half-waves) ---
    const int m = mt * 16 + l16;
    const float* hr = hin + m * DM;
    const float r0 = hr[0], r1 = hr[1], r2 = hr[2],
                r3 = hr[3], r4 = hr[4], r5 = hr[5];
    const float ms = (r0*r0 + r1*r1 + r2*r2 + r3*r3 + r4*r4 + r5*r5)
                     * (1.0f / DM);
    const float s  = __frsqrt_rn(ms + EPS_);
    const float xn0 = r0 * s * normw[0];
    const float xn1 = r1 * s * normw[1];
    const float xn2 = r2 * s * normw[2];
    const float xn3 = r3 * s * normw[3];
    const float xn4 = r4 * s * normw[4];
    const float xn5 = r5 * s * normw[5];

    // --- B-side: full weight row for this lane's column ---
    const int n = nt * 16 + l16;
    const float* wr = w + n * DM;
    const float w0 = wr[0], w1 = wr[1], w2 = wr[2],
                w3 = wr[3], w4 = wr[4], w5 = wr[5];

    // Step 0: K = 0..3
    v2f a0, b0;
    a0.x = hi ? xn2 : xn0;
    a0.y = hi ? xn3 : xn1;
    b0.x = hi ? w2 : w0;
    b0.y = hi ? w3 : w1;
    // Step 1: K = 4..7 (K=6,7 zeroed via B only)
    v2f a1, b1;
    a1.x = xn4;                 // arbitrary on hi lanes (×0)
    a1.y = xn5;
    b1.x = hi ? 0.0f : w4;
    b1.y = hi ? 0.0f : w5;

    v8f acc = {};
    acc = __builtin_amdgcn_wmma_f32_16x16x4_f32(
        false, a0, false, b0, (short)0, acc, false, false);
    acc = __builtin_amdgcn_wmma_f32_16x16x4_f32(
        false, a1, false, b1, (short)0, acc, false, false);

    const int mbase = mt * 16 + (hi ? 8 : 0);
#pragma unroll
    for (int r = 0; r < 8; ++r)
        xz[(mbase + r) * XZ_N + n] = acc[r];
}

// ---------------------------------------------------------------------------
// K2: depthwise causal conv (DCONV=16) + bias + SiLU.
// Branch-free: clamped load address, weight masked by one cndmask.
// ---------------------------------------------------------------------------
__global__ void k_conv(const float* __restrict__ xz,  // x_in = cols [0,48)
                       const float* __restrict__ cw,  // (48,16)
                       const float* __restrict__ cb,  // (48)
                       float* __restrict__ xc)        // (M,48)
{
    const int idx = blockIdx.x * blockDim.x + threadIdx.x;
    if (idx >= MTOT * ED) return;
    const int e = idx % ED;
    const int m = idx / ED;
    const int l = m % L_SZ;

    float acc = cb[e];
    const float* wrow = cw + e * DCONV;
#pragma unroll
    for (int j = 0; j < DCONV; ++j) {
        const int ls = l - (DCONV - 1) + j;
        const int mg = (ls >= 0) ? (m - (DCONV - 1) + j) : m;  // clamped addr
        const float wj = (ls >= 0) ? wrow[j] : 0.0f;            // 1 cndmask
        acc = fmaf(xz[mg * XZ_N + e], wj, acc);
    }
    xc[idx] = acc * fast_sigmoid(acc);
}

// ---------------------------------------------------------------------------
// K3: x_proj.  dbl[m, 0:65] = x_c[m,:] @ x_proj_w.T   (N padded to 80)
// GEMM M=16384 N=65 K=48 via wmma f32 16x16x4 (12 K-steps).
// Columns 65..79 use a CLAMPED weight row (dup of col 64); they are padding
// that nothing downstream reads, so no masking is required at all.
// ---------------------------------------------------------------------------
__global__ void k_xproj(const float* __restrict__ xc,  // (M,48)
                        const float* __restrict__ w,   // (65,48)
                        float* __restrict__ dbl)       // (M,80)
{
    const int gw   = blockIdx.x * 4 + (threadIdx.x >> 5);
    const int mt   = gw / 5;
    const int nt   = gw % 5;
    const int t    = threadIdx.x & 31;
    const int l16  = t & 15;
    const int half = t >> 4;
    const int kh   = half * 2;

    const int m  = mt * 16 + l16;
    const int n  = nt * 16 + l16;
    const int nc = (n < 65) ? n : 64;       // clamp (padding cols never read)
    const float* arow = xc + m * ED + kh;   // lane-dependent base
    const float* brow = w + nc * ED + kh;

    v8f acc = {};
#pragma unroll
    for (int ks = 0; ks < 12; ++ks) {
        v2f a, b;
        a.x = arow[ks * 4];
        a.y = arow[ks * 4 + 1];
        b.x = brow[ks * 4];
        b.y = brow[ks * 4 + 1];
        acc = __builtin_amdgcn_wmma_f32_16x16x4_f32(
            false, a, false, b, (short)0, acc, false, false);
    }

    const int mbase = mt * 16 + half * 8;
#pragma unroll
    for (int r = 0; r < 8; ++r)
        dbl[(mbase + r) * DBL_N + n] = acc[r];
}

// ---------------------------------------------------------------------------
// K3b: delta = softplus(dr * dt_w[e] + dt_b[e])   (DTRANK = 1)
// ---------------------------------------------------------------------------
__global__ void k_delta(const float* __restrict__ dbl, // (M,80), col 0 = dr
                        const float* __restrict__ dtw, // (48)
                        const float* __restrict__ dtb, // (48)
                        float* __restrict__ delta)     // (M,48)
{
    const int idx = blockIdx.x * blockDim.x + threadIdx.x;
    if (idx >= MTOT * ED) return;
    const int e = idx % ED;
    const int m = idx / ED;
    const float v = fmaf(dbl[m * DBL_N], dtw[e], dtb[e]);
    delta[idx] = (v > 20.0f) ? v : log1pf(__expf(v));
}

// ---------------------------------------------------------------------------
// K4: SSM scan + D skip + SiLU(z) gate.
// One wave per (b, e); lane = state index n (wave32 == NST exactly).
// grid = B * (ED/8) = 48 blocks of 256 threads (8 waves).
// ---------------------------------------------------------------------------
__global__ void k_scan(const float* __restrict__ dbl,   // (M,80): B=1..32, C=33..64
                       const float* __restrict__ delta, // (M,48)
                       const float* __restrict__ xc,    // (M,48)
                       const float* __restrict__ xz,    // z = cols [48,96)
                       const float* __restrict__ Alog,  // (48,32)
                       const float* __restrict__ Dp,    // (48)
                       float* __restrict__ y)           // (M,48)
{
    const int b    = blockIdx.x / 6;
    const int eg   = blockIdx.x % 6;
    const int wv   = threadIdx.x >> 5;
    const int lane = threadIdx.x & 31;
    const int e    = eg * 8 + wv;

    const float Aval = -__expf(Alog[e * NST + lane]);
    const float Dv   = Dp[e];
    const int base   = b * L_SZ;

    float h = 0.0f;
    for (int l = 0; l < L_SZ; ++l) {
        const int m = base + l;
        const float dv = delta[m * ED + e];
        const float xv = xc[m * ED + e];
        const float Bv = dbl[m * DBL_N + 1 + lane];
        const float Cv = dbl[m * DBL_N + 33 + lane];

        const float dA = __expf(dv * Aval);
        h = fmaf(dA, h, dv * xv * Bv);

        float p = h * Cv;
        p += __shfl_xor(p, 16, 32);
        p += __shfl_xor(p, 8, 32);
        p += __shfl_xor(p, 4, 32);
        p += __shfl_xor(p, 2, 32);
        p += __shfl_xor(p, 1, 32);

        if (lane == 0) {
            float yv = p + Dv * xv;
            const float zv = xz[m * XZ_N + ED + e];
            yv *= zv * fast_sigmoid(zv);
            y[m * ED + e] = yv;
        }
    }
}

// ---------------------------------------------------------------------------
// K5: out_proj + residual.  h[m,0:6] = y[m,:] @ out_w.T + h[m,0:6]
// GEMM M=16384 N=6(pad 16) K=48; residual seeded into the C accumulator.
// Invalid columns (6..15) use clamped addresses and are simply never stored.
// ---------------------------------------------------------------------------
__global__ void k_outproj(const float* __restrict__ y,    // (M,48)
                          const float* __restrict__ w,    // (6,48)
                          const float* __restrict__ hres, // (M,6)
                          float* __restrict__ hout)       // (M,6)
{
    const int mt   = blockIdx.x * 4 + (threadIdx.x >> 5);
    const int t    = threadIdx.x & 31;
    const int l16  = t & 15;
    const int half = t >> 4;
    const int kh   = half * 2;

    const int m = mt * 16 + l16;
    const int n = l16;                     // single N tile, cols 0..5 valid
    const bool nvalid = (n < DM);
    const int nc = nvalid ? n : 0;         // clamp; garbage cols never stored
    const float* arow = y + m * ED + kh;
    const float* brow = w + nc * ED + kh;

    const int mbase = mt * 16 + half * 8;

    v8f acc;
#pragma unroll
    for (int r = 0; r < 8; ++r)
        acc[r] = hres[(mbase + r) * DM + nc];   // seed C with residual

#pragma unroll
    for (int ks = 0; ks < 12; ++ks) {
        v2f a, b;
        a.x = arow[ks * 4];
        a.y = arow[ks * 4 + 1];
        b.x = brow[ks * 4];
        b.y = brow[ks * 4 + 1];
        acc = __builtin_amdgcn_wmma_f32_16x16x4_f32(
            false, a, false, b, (short)0, acc, false, false);
    }

    if (nvalid) {
#pragma unroll
        for (int r = 0; r < 8; ++r)
            hout[(mbase + r) * DM + n] = acc[r];
    }
}

// ---------------------------------------------------------------------------
// K6: classifier head.  out[b,c] = y[b, L-1, :] @ fc_w.T + fc_b   (8x4, K=48)
// ---------------------------------------------------------------------------
__global__ void k_final(const float* __restrict__ y,   // (M,48), last layer's y
                        const float* __restrict__ fcw, // (4,48)
                        const float* __restrict__ fcb, // (4)
                        float* __restrict__ out)       // (8,4)
{
    const int t = threadIdx.x;
    if (t >= 32) return;
    const int bq = t >> 2;
    const int c  = t & 3;
    const float* yr = y + (bq * L_SZ + (L_SZ - 1)) * ED;
    float acc = fcb[c];
#pragma unroll
    for (int k = 0; k < ED; ++k)
        acc = fmaf(yr[k], fcw[c * ED + k], acc);
    out[bq * 4 + c] = acc;
}

// ---------------------------------------------------------------------------
// Host side
// ---------------------------------------------------------------------------
extern "C" void kernel_launch(void* const* d_in, const int* in_sizes, int n_in,
                              void* d_out, int out_size, void* d_ws, size_t ws_size,
                              hipStream_t stream) {
    const float* x      = (const float*)d_in[0];   // (8,2048,6)
    const float* inpw   = (const float*)d_in[1];   // (4,96,6)
    const float* convw  = (const float*)d_in[2];   // (4,48,16)
    const float* convb  = (const float*)d_in[3];   // (4,48)
    const float* xprojw = (const float*)d_in[4];   // (4,65,48)
    const float* dtw    = (const float*)d_in[5];   // (4,48,1)
    const float* dtb    = (const float*)d_in[6];   // (4,48)
    const float* Alog   = (const float*)d_in[7];   // (4,48,32)
    const float* Dp     = (const float*)d_in[8];   // (4,48)
    const float* outw   = (const float*)d_in[9];   // (4,6,48)
    const float* normw  = (const float*)d_in[10];  // (4,6)
    const float* fcw    = (const float*)d_in[11];  // (4,48)
    const float* fcb    = (const float*)d_in[12];  // (4)
    float* out          = (float*)d_out;           // (8,4)

    float* ws    = (float*)d_ws;
    float* xz    = ws;                           // M * 96
    float* xc    = xz    + (size_t)MTOT * XZ_N;  // M * 48
    float* dbl   = xc    + (size_t)MTOT * ED;    // M * 80
    float* delta = dbl   + (size_t)MTOT * DBL_N; // M * 48
    float* ybuf  = delta + (size_t)MTOT * ED;    // M * 48
    float* hbuf  = ybuf  + (size_t)MTOT * ED;    // M * 6

    const int NLAYERS = 4;
    for (int i = 0; i < NLAYERS; ++i) {
        const float* hin = (i == 0) ? x : hbuf;

        // K1: rmsnorm + in_proj  (1536 blocks x 128 thr = 6144 waves)
        k_norm_inproj<<<(MT * 6) / 4, 128, 0, stream>>>(
            hin, inpw + (size_t)i * 96 * DM, normw + (size_t)i * DM, xz);

        // K2: conv + silu
        k_conv<<<(MTOT * ED + 255) / 256, 256, 0, stream>>>(
            xz, convw + (size_t)i * ED * DCONV, convb + (size_t)i * ED, xc);

        // K3: x_proj  (1280 blocks x 128 thr = 5120 waves)
        k_xproj<<<(MT * 5) / 4, 128, 0, stream>>>(
            xc, xprojw + (size_t)i * 65 * ED, dbl);

        // K3b: delta = softplus
        k_delta<<<(MTOT * ED + 255) / 256, 256, 0, stream>>>(
            dbl, dtw + (size_t)i * ED, dtb + (size_t)i * ED, delta);

        // K4: sequential SSM scan, one wave per (b,e)
        k_scan<<<B_SZ * (ED / 8), 256, 0, stream>>>(
            dbl, delta, xc, xz,
            Alog + (size_t)i * ED * NST, Dp + (size_t)i * ED, ybuf);

        if (i < NLAYERS - 1) {
            // K5: out_proj + residual (in-place on hbuf for i>=1)
            k_outproj<<<MT / 4, 128, 0, stream>>>(
                ybuf, outw + (size_t)i * DM * ED, hin, hbuf);
        }
    }

    // K6: classifier head on last layer's gated y
    k_final<<<1, 32, 0, stream>>>(ybuf, fcw, fcb, out);
}